// ScaleNet_2025_43447889166751
// MI455X (gfx1250) — compile-verified
//
#include <hip/hip_runtime.h>
#include <hip/hip_bf16.h>

typedef __attribute__((ext_vector_type(2))) float v2f;
typedef __attribute__((ext_vector_type(8))) float v8f;

#define D 32  // D_IN == D_OUT == 32

// ---------------------------------------------------------------- utilities
__global__ void gcn_zero_f32(float* __restrict__ p, int n) {
    int i = blockIdx.x * blockDim.x + threadIdx.x;
    if (i < n) p[i] = 0.0f;
}

// out[n, d] = bias[d]   (also serves as the "+ bias" term; atomics accumulate on top)
__global__ void gcn_init_out(float* __restrict__ out, const float* __restrict__ bias, int total) {
    int i = blockIdx.x * blockDim.x + threadIdx.x;
    if (i < total) out[i] = bias[i & (D - 1)];
}

// deg[col[e]] += w[e]
__global__ void gcn_degree(const int* __restrict__ ei, const float* __restrict__ ew,
                           float* __restrict__ deg, int nE) {
    int e = blockIdx.x * blockDim.x + threadIdx.x;
    if (e >= nE) return;
    int col = ei[nE + e];           // edge_index[1][e]
    atomicAdd(&deg[col], ew[e]);
}

// dis[i] = deg[i] > 0 ? rsqrt(deg[i]) : 0
__global__ void gcn_inv_sqrt(const float* __restrict__ deg, float* __restrict__ dis, int n) {
    int i = blockIdx.x * blockDim.x + threadIdx.x;
    if (i >= n) return;
    float d = deg[i];
    dis[i] = (d > 0.0f) ? rsqrtf(d) : 0.0f;
}

// ---------------------------------------------------------------- h = x @ W  (WMMA fp32)
// One wave -> one 16x32 tile of h. A: 16x4 f32 fragments (2 VGPRs),
// B/C/D in the CDNA5 row-striped layout. 8 K-steps x 2 N-tiles = 16 WMMAs.
__global__ void gcn_gemm_wmma(const float* __restrict__ x, const float* __restrict__ Wm,
                              float* __restrict__ h, int n) {
    const int lane  = threadIdx.x & 31;
    const int wave  = threadIdx.x >> 5;
    const int tile  = blockIdx.x * (blockDim.x >> 5) + wave;
    const int row0  = tile << 4;
    if (row0 >= n) return;                 // wave-uniform exit

    const int m  = lane & 15;              // M (for A) / N (for B,C,D)
    const int kh = lane >> 4;              // which half of the K-quad / M-half for C

    if (row0 + 16 <= n) {                  // full tile: WMMA path (EXEC all 1s)
        v8f c0 = {};                       // output cols 0..15
        v8f c1 = {};                       // output cols 16..31
        const float* xr = x + (size_t)(row0 + m) * D;
#pragma unroll
        for (int j = 0; j < 8; ++j) {
            const int kb = 4 * j + 2 * kh; // A layout: v0 -> K=kb, v1 -> K=kb+1
            v2f a;
            a[0] = xr[kb];
            a[1] = xr[kb + 1];
            v2f b0, b1;                    // B layout: row K striped across lanes
            b0[0] = Wm[kb * D + m];
            b0[1] = Wm[(kb + 1) * D + m];
            b1[0] = Wm[kb * D + 16 + m];
            b1[1] = Wm[(kb + 1) * D + 16 + m];
            c0 = __builtin_amdgcn_wmma_f32_16x16x4_f32(false, a, false, b0,
                                                       (short)0, c0, false, false);
            c1 = __builtin_amdgcn_wmma_f32_16x16x4_f32(false, a, false, b1,
                                                       (short)0, c1, false, false);
        }
        // C/D layout: VGPR r holds M = r (lanes 0-15) / M = 8+r (lanes 16-31), N = m
#pragma unroll
        for (int r = 0; r < 8; ++r) {
            const int mr = row0 + r + 8 * kh;
            h[(size_t)mr * D + m]      = c0[r];
            h[(size_t)mr * D + 16 + m] = c1[r];
        }
    } else {
        // scalar tail (never taken for N=100000, kept for generality)
        for (int r = row0 + lane; r < n; r += 32) {
            for (int dc = 0; dc < D; ++dc) {
                float acc = 0.0f;
                for (int k = 0; k < D; ++k) acc += x[(size_t)r * D + k] * Wm[k * D + dc];
                h[(size_t)r * D + dc] = acc;
            }
        }
    }
}

// ---------------------------------------------------------------- edge scatter
// thread = (edge, channel). Lanes of one edge read h[row*32 .. +31] (coalesced 128B)
// and atomically add into out[col*32 .. +31] (coalesced atomics, L2-resident).
__global__ void gcn_scatter(const int* __restrict__ ei, const float* __restrict__ ew,
                            const float* __restrict__ dis, const float* __restrict__ h,
                            float* __restrict__ out, int nE) {
    long long tid = (long long)blockIdx.x * blockDim.x + threadIdx.x;
    int e = (int)(tid >> 5);
    int d = (int)(tid & 31);
    if (e >= nE) return;
    int row = ei[e];                 // source
    int col = ei[nE + e];            // target
    float nrm = dis[row] * ew[e] * dis[col];
    float msg = nrm * h[(size_t)row * D + d];
    atomicAdd(&out[(size_t)col * D + d], msg);
}

// ---------------------------------------------------------------- launcher
extern "C" void kernel_launch(void* const* d_in, const int* in_sizes, int n_in,
                              void* d_out, int out_size, void* d_ws, size_t ws_size,
                              hipStream_t stream) {
    const float* x    = (const float*)d_in[0];   // [N, 32]
    const int*   ei   = (const int*)d_in[1];     // [2, E]
    const float* ew   = (const float*)d_in[2];   // [E]
    const float* Wm   = (const float*)d_in[3];   // [32, 32]
    const float* bias = (const float*)d_in[4];   // [32]
    float* out = (float*)d_out;                  // [N, 32]

    const int n = in_sizes[0] / D;               // 100000
    const int e = in_sizes[2];                   // 1600000

    float* deg = (float*)d_ws;                   // [N]
    float* dis = deg + n;                        // [N]
    float* h   = dis + n;                        // [N, 32]

    const int B = 256;

    // 1) deg = 0
    gcn_zero_f32<<<(n + B - 1) / B, B, 0, stream>>>(deg, n);
    // 2) deg[col] += w
    gcn_degree<<<(e + B - 1) / B, B, 0, stream>>>(ei, ew, deg, e);
    // 3) dis = deg>0 ? rsqrt(deg) : 0
    gcn_inv_sqrt<<<(n + B - 1) / B, B, 0, stream>>>(deg, dis, n);
    // 4) h = x @ W   (WMMA, 8 waves/block -> 128 rows/block)
    {
        int tiles  = (n + 15) / 16;
        int wavesPerBlock = B / 32;
        int blocks = (tiles + wavesPerBlock - 1) / wavesPerBlock;
        gcn_gemm_wmma<<<blocks, B, 0, stream>>>(x, Wm, h, n);
    }
    // 5) out = bias (broadcast)
    gcn_init_out<<<(n * D + B - 1) / B, B, 0, stream>>>(out, bias, n * D);
    // 6) out[col] += (dis[row]*w*dis[col]) * h[row]
    {
        long long total = (long long)e * D;
        int blocks = (int)((total + B - 1) / B);
        gcn_scatter<<<blocks, B, 0, stream>>>(ei, ew, dis, h, out, e);
    }
}